// RGCNLayer_85993835200926
// MI455X (gfx1250) — compile-verified
//
#include <hip/hip_runtime.h>

typedef float v2f __attribute__((ext_vector_type(2)));
typedef float v8f __attribute__((ext_vector_type(8)));

#define NNODES 50000
#define DIN 64
#define DOUT 64
#define NREL 16
#define SCAN_BLK 256
#define NBLK ((NNODES + SCAN_BLK - 1) / SCAN_BLK)   // 196

// ---------------------------------------------------------------------------
// Kernel 1: T[r, n, :] = X[n, :] @ W[r]   (fp32 WMMA, exact precision)
// grid: (N/16, R), block: 128 threads = 4 waves; wave w -> cols [16w,16w+16)
// ---------------------------------------------------------------------------
__global__ __launch_bounds__(128)
void rgcn_transform(const float* __restrict__ X,
                    const float* __restrict__ W,
                    float* __restrict__ T) {
  __shared__ float ldsX[16 * DIN];     // 4 KB
  __shared__ float ldsW[DIN * DOUT];   // 16 KB
  const int tid  = threadIdx.x;
  const int row0 = blockIdx.x * 16;
  const int r    = blockIdx.y;

  {
    const float4* src = (const float4*)(X + (size_t)row0 * DIN);
    float4* dst = (float4*)ldsX;
#pragma unroll
    for (int i = 0; i < 2; ++i) dst[tid + i * 128] = src[tid + i * 128];
  }
  {
    const float4* src = (const float4*)(W + (size_t)r * DIN * DOUT);
    float4* dst = (float4*)ldsW;
#pragma unroll
    for (int i = 0; i < 8; ++i) dst[tid + i * 128] = src[tid + i * 128];
  }
  __syncthreads();

  const int lane = tid & 31;
  const int c0   = (tid >> 5) * 16;
  const int half = lane >> 4;
  const int l15  = lane & 15;

  v8f acc = {};
#pragma unroll
  for (int j = 0; j < 16; ++j) {
    const int k = j * 4 + half * 2;
    v2f a = *(const v2f*)(ldsX + l15 * DIN + k);
    v2f b;
    b.x = ldsW[(k + 0) * DOUT + c0 + l15];
    b.y = ldsW[(k + 1) * DOUT + c0 + l15];
    acc = __builtin_amdgcn_wmma_f32_16x16x4_f32(false, a, false, b,
                                                (short)0, acc, false, false);
  }

  float* out = T + ((size_t)r * NNODES + row0) * DOUT + c0;
#pragma unroll
  for (int v = 0; v < 8; ++v)
    out[(size_t)(v + 8 * half) * DOUT + l15] = acc[v];
}

// ---------------------------------------------------------------------------
// Kernel 2: histogram of edge destinations (1 u32 atomic per edge)
// ---------------------------------------------------------------------------
__global__ __launch_bounds__(256)
void rgcn_hist(const int* __restrict__ edst, unsigned* __restrict__ counts,
               int E) {
  int e = blockIdx.x * blockDim.x + threadIdx.x;
  if (e < E) atomicAdd(&counts[edst[e]], 1u);
}

// ---------------------------------------------------------------------------
// Scan step A: per-block sums of counts
// ---------------------------------------------------------------------------
__global__ __launch_bounds__(SCAN_BLK)
void rgcn_scanA(const unsigned* __restrict__ counts,
                unsigned* __restrict__ blockSums) {
  __shared__ unsigned s[SCAN_BLK];
  int i = blockIdx.x * SCAN_BLK + threadIdx.x;
  s[threadIdx.x] = (i < NNODES) ? counts[i] : 0u;
  __syncthreads();
  for (int off = SCAN_BLK >> 1; off > 0; off >>= 1) {
    if (threadIdx.x < off) s[threadIdx.x] += s[threadIdx.x + off];
    __syncthreads();
  }
  if (threadIdx.x == 0) blockSums[blockIdx.x] = s[0];
}

// ---------------------------------------------------------------------------
// Scan step B: single-block exclusive scan of block sums (NBLK <= 256)
// ---------------------------------------------------------------------------
__global__ __launch_bounds__(SCAN_BLK)
void rgcn_scanB(unsigned* __restrict__ blockSums,
                unsigned* __restrict__ blockBase) {
  __shared__ unsigned s[SCAN_BLK];
  unsigned v = (threadIdx.x < NBLK) ? blockSums[threadIdx.x] : 0u;
  s[threadIdx.x] = v;
  __syncthreads();
  for (int off = 1; off < SCAN_BLK; off <<= 1) {
    unsigned t = (threadIdx.x >= off) ? s[threadIdx.x - off] : 0u;
    __syncthreads();
    s[threadIdx.x] += t;
    __syncthreads();
  }
  if (threadIdx.x < NBLK) blockBase[threadIdx.x] = s[threadIdx.x] - v;
}

// ---------------------------------------------------------------------------
// Scan step C: exclusive scan within block + base -> offsets, cursor
// ---------------------------------------------------------------------------
__global__ __launch_bounds__(SCAN_BLK)
void rgcn_scanC(const unsigned* __restrict__ counts,
                const unsigned* __restrict__ blockBase,
                unsigned* __restrict__ offsets,
                unsigned* __restrict__ cursor) {
  __shared__ unsigned s[SCAN_BLK];
  int i = blockIdx.x * SCAN_BLK + threadIdx.x;
  unsigned v = (i < NNODES) ? counts[i] : 0u;
  s[threadIdx.x] = v;
  __syncthreads();
  for (int off = 1; off < SCAN_BLK; off <<= 1) {
    unsigned t = (threadIdx.x >= off) ? s[threadIdx.x - off] : 0u;
    __syncthreads();
    s[threadIdx.x] += t;
    __syncthreads();
  }
  if (i < NNODES) {
    unsigned o = s[threadIdx.x] - v + blockBase[blockIdx.x];
    offsets[i] = o;
    cursor[i]  = o;
  }
}

// ---------------------------------------------------------------------------
// Kernel 3: bucket edges by destination; payload = (src | type<<16, norm)
// ---------------------------------------------------------------------------
__global__ __launch_bounds__(256)
void rgcn_bucket(const int* __restrict__ esrc, const int* __restrict__ edst,
                 const int* __restrict__ etype, const float* __restrict__ enorm,
                 unsigned* __restrict__ cursor,
                 unsigned long long* __restrict__ sorted, int E) {
  int e = blockIdx.x * blockDim.x + threadIdx.x;
  if (e >= E) return;
  unsigned pos = atomicAdd(&cursor[edst[e]], 1u);
  unsigned key = (unsigned)esrc[e] | ((unsigned)etype[e] << 16);
  unsigned long long p =
      (unsigned long long)key |
      ((unsigned long long)__float_as_uint(enorm[e]) << 32);
  sorted[pos] = p;
}

// ---------------------------------------------------------------------------
// Kernel 4: per-node accumulation + bias + ReLU. One wave per node.
// No fp atomics: registers accumulate ~20 gathered rows, one coalesced store.
// ---------------------------------------------------------------------------
__global__ __launch_bounds__(256)
void rgcn_aggregate(const float* __restrict__ T,
                    const unsigned long long* __restrict__ sorted,
                    const unsigned* __restrict__ offsets,
                    const unsigned* __restrict__ counts,
                    const float* __restrict__ bias,
                    float* __restrict__ out) {
  const int n = blockIdx.x * (blockDim.x >> 5) + (threadIdx.x >> 5);
  if (n >= NNODES) return;
  const int lane = threadIdx.x & 31;
  const unsigned start = offsets[n];
  const unsigned len   = counts[n];

  v2f acc = {0.0f, 0.0f};
  for (unsigned i = 0; i < len; ++i) {
    unsigned long long p = sorted[start + i];
    unsigned key = (unsigned)p;
    float nrm = __uint_as_float((unsigned)(p >> 32));
    int s = key & 0xFFFF;
    int t = key >> 16;
    const v2f* row = (const v2f*)(T + ((size_t)t * NNODES + s) * DOUT);
    v2f v = row[lane];                 // coalesced 256B gather per wave
    acc.x += v.x * nrm;
    acc.y += v.y * nrm;
  }
  v2f b = *(const v2f*)(bias + lane * 2);
  v2f o;
  o.x = fmaxf(acc.x + b.x, 0.0f);
  o.y = fmaxf(acc.y + b.y, 0.0f);
  *(v2f*)(out + (size_t)n * DOUT + lane * 2) = o;
}

// ---------------------------------------------------------------------------
extern "C" void kernel_launch(void* const* d_in, const int* in_sizes, int n_in,
                              void* d_out, int out_size, void* d_ws,
                              size_t ws_size, hipStream_t stream) {
  const float* X     = (const float*)d_in[0];
  const int*   esrc  = (const int*)d_in[1];
  const int*   edst  = (const int*)d_in[2];
  const int*   etype = (const int*)d_in[3];
  const float* enorm = (const float*)d_in[4];
  const float* W     = (const float*)d_in[5];
  const float* bias  = (const float*)d_in[6];
  float* out = (float*)d_out;
  const int E = in_sizes[1];

  // Workspace layout (256B-aligned regions)
  char* base = (char*)d_ws;
  const size_t TBYTES = (size_t)NREL * NNODES * DOUT * sizeof(float); // 204.8MB
  float*              T         = (float*)base;
  size_t off = TBYTES;
  auto align256 = [](size_t x) { return (x + 255) & ~(size_t)255; };
  unsigned* counts    = (unsigned*)(base + off); off = align256(off + (NNODES + 1) * 4);
  unsigned* offsets   = (unsigned*)(base + off); off = align256(off + (NNODES + 1) * 4);
  unsigned* cursor    = (unsigned*)(base + off); off = align256(off + (NNODES + 1) * 4);
  unsigned* blockSums = (unsigned*)(base + off); off = align256(off + SCAN_BLK * 4);
  unsigned* blockBase = (unsigned*)(base + off); off = align256(off + SCAN_BLK * 4);
  unsigned long long* sorted = (unsigned long long*)(base + off);

  // Phase 0: zero histogram bins
  hipMemsetAsync(counts, 0, (NNODES + 1) * sizeof(unsigned), stream);

  // Phase 1: dense WMMA transform  T[r] = X @ W[r]
  dim3 g1(NNODES / 16, NREL);
  rgcn_transform<<<g1, 128, 0, stream>>>(X, W, T);

  // Phase 2: counting sort of edges by destination
  const int eblk = (E + 255) / 256;
  rgcn_hist<<<eblk, 256, 0, stream>>>(edst, counts, E);
  rgcn_scanA<<<NBLK, SCAN_BLK, 0, stream>>>(counts, blockSums);
  rgcn_scanB<<<1, SCAN_BLK, 0, stream>>>(blockSums, blockBase);
  rgcn_scanC<<<NBLK, SCAN_BLK, 0, stream>>>(counts, blockBase, offsets, cursor);
  rgcn_bucket<<<eblk, 256, 0, stream>>>(esrc, edst, etype, enorm, cursor,
                                        sorted, E);

  // Phase 3: per-node gather-accumulate + bias + ReLU (wave per node)
  const int nblk = (NNODES + 7) / 8;
  rgcn_aggregate<<<nblk, 256, 0, stream>>>(T, sorted, offsets, counts, bias,
                                           out);
}